// RefSliceSoftSort_52080773431755
// MI455X (gfx1250) — compile-verified
//
#include <hip/hip_runtime.h>
#include <stdint.h>

// Problem constants (fixed by the reference)
#define N_COLS   131072
#define K_TOP    4096
#define CAP      8192        // candidate capacity (power of 2 for bitonic; holds ties at threshold)
#define TPB_A    1024        // 32 waves (wave32) for the per-row top-k block
#define CHUNK    2048        // columns per block in the nearest-neighbor kernel
#define TPB_B    256

// ---------------------------------------------------------------------------
// Orderable-uint transform: monotonic float -> uint32 mapping
// ---------------------------------------------------------------------------
__device__ __forceinline__ unsigned int ord_u32(float f) {
  unsigned int u = __float_as_uint(f);
  return (u & 0x80000000u) ? ~u : (u | 0x80000000u);
}
__device__ __forceinline__ float inv_ord(unsigned int u) {
  unsigned int v = (u & 0x80000000u) ? (u ^ 0x80000000u) : ~u;
  return __uint_as_float(v);
}

// ---------------------------------------------------------------------------
// Kernel A: per-row exact top-K (sorted desc, stable) via radix select +
// LDS bitonic sort. One workgroup per row.
// Dynamic LDS layout: [CAP x u64 candidates][256 x u32 hist][4 x u32 ctrl]
// ---------------------------------------------------------------------------
__global__ __launch_bounds__(TPB_A)
void topk_radix_sort_kernel(const float* __restrict__ scores,
                            float* __restrict__ ref_val,
                            int* __restrict__ ref_idx) {
  extern __shared__ unsigned char s_raw[];
  unsigned long long* cand = (unsigned long long*)s_raw;
  unsigned int* hist = (unsigned int*)(s_raw + (size_t)CAP * 8);
  unsigned int* ctrl = (unsigned int*)(s_raw + (size_t)CAP * 8 + 256 * 4);
  // ctrl[0]=prefix, ctrl[1]=k_remaining, ctrl[2]=candidate count

  const int row = blockIdx.x;
  const int tid = threadIdx.x;
  const int nt  = blockDim.x;
  const float4* rp4 = (const float4*)(scores + (size_t)row * N_COLS);

  if (tid == 0) { ctrl[0] = 0u; ctrl[1] = K_TOP; ctrl[2] = 0u; }
  __syncthreads();

  // ---- 4-pass MSB-first radix select: find exact value of K-th largest ----
  for (int shift = 24; shift >= 0; shift -= 8) {
    for (int i = tid; i < 256; i += nt) hist[i] = 0u;
    __syncthreads();
    const unsigned int pfx   = ctrl[0];
    const unsigned int hmask = (shift == 24) ? 0u : (0xFFFFFFFFu << (shift + 8));
    for (int i = tid; i < N_COLS / 4; i += nt) {
      float4 v = rp4[i];
      float vals[4] = {v.x, v.y, v.z, v.w};
#pragma unroll
      for (int e = 0; e < 4; ++e) {
        unsigned int u = ord_u32(vals[e]);
        if ((u & hmask) == pfx) atomicAdd(&hist[(u >> shift) & 255u], 1u);
      }
    }
    __syncthreads();
    if (tid == 0) {
      unsigned int kk = ctrl[1], cum = 0u;
      for (int b = 255; b >= 0; --b) {
        unsigned int h = hist[b];
        if (cum + h >= kk) {
          ctrl[1] = kk - cum;
          ctrl[0] = pfx | ((unsigned int)b << shift);
          break;
        }
        cum += h;
      }
    }
    __syncthreads();
  }
  const unsigned int T = ctrl[0];   // exact orderable value of the K-th largest

  // ---- compact all candidates u >= T into LDS as {u, ~idx} keys ----------
  for (int i = tid; i < N_COLS / 4; i += nt) {
    float4 v = rp4[i];
    float vals[4] = {v.x, v.y, v.z, v.w};
#pragma unroll
    for (int e = 0; e < 4; ++e) {
      unsigned int u = ord_u32(vals[e]);
      if (u >= T) {
        unsigned int p = atomicAdd(&ctrl[2], 1u);
        if (p < CAP) {
          unsigned int idx = (unsigned int)(i * 4 + e);
          cand[p] = ((unsigned long long)u << 32) |
                    (unsigned long long)(0xFFFFFFFFu - idx);
        }
      }
    }
  }
  __syncthreads();
  unsigned int cnt = ctrl[2];
  if (cnt > CAP) cnt = CAP;
  for (unsigned int i = cnt + tid; i < CAP; i += nt) cand[i] = 0ull;  // pad: sorts last
  __syncthreads();

  // ---- bitonic sort CAP keys descending (u desc, idx asc on ties) --------
  for (unsigned int k2 = 2; k2 <= CAP; k2 <<= 1) {
    for (unsigned int j = k2 >> 1; j > 0; j >>= 1) {
      for (unsigned int i = tid; i < CAP; i += nt) {
        unsigned int ixj = i ^ j;
        if (ixj > i) {
          bool desc = ((i & k2) == 0);
          unsigned long long a = cand[i];
          unsigned long long b = cand[ixj];
          if (desc ? (a < b) : (a > b)) { cand[i] = b; cand[ixj] = a; }
        }
      }
      __syncthreads();
    }
  }

  // ---- emit sorted top-K values (as float) and original indices ----------
  for (int r = tid; r < K_TOP; r += nt) {
    unsigned long long key = cand[r];
    unsigned int u   = (unsigned int)(key >> 32);
    unsigned int idx = 0xFFFFFFFFu - (unsigned int)(key & 0xFFFFFFFFu);
    ref_val[(size_t)row * K_TOP + r] = inv_ord(u);
    ref_idx[(size_t)row * K_TOP + r] = (int)idx;
  }
}

// ---------------------------------------------------------------------------
// Kernel B: nearest-ref binary search + index gather.
// Stages the row's 32KB ref tables into LDS (async global->LDS when the
// CDNA5 builtins are available), prefetches the score chunk, then does a
// fully unrolled 12-step descending binary search per element.
// ---------------------------------------------------------------------------
#if defined(__AMDGCN__) && \
    __has_builtin(__builtin_amdgcn_global_load_async_to_lds_b128) && \
    __has_builtin(__builtin_amdgcn_s_wait_asynccnt)
#define USE_ASYNC_LDS 1
// Builtin signature (from the gfx1250 clang diagnostic):
//   void __builtin_amdgcn_global_load_async_to_lds_b128(
//       v4i addrspace(1)* src, v4i addrspace(3)* dst, imm int off, imm int cpol)
typedef int v4i_t __attribute__((vector_size(16)));
typedef v4i_t __attribute__((address_space(1))) gv4i_t;   // global (__device__)
typedef v4i_t __attribute__((address_space(3))) lv4i_t;   // LDS (__shared__)
#endif

__global__ __launch_bounds__(TPB_B)
void nearest_gather_kernel(const float* __restrict__ scores,
                           const float* __restrict__ ref_val,
                           const int* __restrict__ ref_idx,
                           int* __restrict__ out) {
  __shared__ float s_val[K_TOP];
  __shared__ int   s_idx[K_TOP];

  const int row   = blockIdx.y;
  const int tid   = threadIdx.x;
  const int base  = blockIdx.x * CHUNK;
  const float* rv   = ref_val + (size_t)row * K_TOP;
  const int*   ri   = ref_idx + (size_t)row * K_TOP;
  const float* rowp = scores + (size_t)row * N_COLS + base;
  int*         orow = out    + (size_t)row * N_COLS + base;

#ifdef USE_ASYNC_LDS
  // CDNA5 async global->LDS DMA: 16B per lane per issue, tracked w/ ASYNCcnt.
  for (int i = tid; i < K_TOP / 4; i += TPB_B) {
    __builtin_amdgcn_global_load_async_to_lds_b128(
        (gv4i_t*)(void*)(rv + i * 4), (lv4i_t*)(void*)(&s_val[i * 4]), 0, 0);
    __builtin_amdgcn_global_load_async_to_lds_b128(
        (gv4i_t*)(void*)(ri + i * 4), (lv4i_t*)(void*)(&s_idx[i * 4]), 0, 0);
  }
  // overlap: prefetch this block's score chunk while the DMA runs
  for (int i = tid; i < CHUNK; i += TPB_B) __builtin_prefetch(rowp + i, 0, 0);
  __builtin_amdgcn_s_wait_asynccnt(0);
  __syncthreads();
#else
  for (int i = tid; i < CHUNK; i += TPB_B) __builtin_prefetch(rowp + i, 0, 0);
  for (int i = tid; i < K_TOP; i += TPB_B) { s_val[i] = rv[i]; s_idx[i] = ri[i]; }
  __syncthreads();
#endif

  for (int i = tid; i < CHUNK; i += TPB_B) {
    float s = rowp[i];
    // j = count of ref values strictly greater than s (array is descending)
    int j = 0;
#pragma unroll
    for (int w = K_TOP >> 1; w > 0; w >>= 1)
      if (s_val[j + w - 1] > s) j += w;
    if (j < K_TOP && s_val[j] > s) ++j;
    // nearest of {j-1, j}; tie -> smaller index (argmax-first semantics)
    int pick;
    if (j <= 0)            pick = 0;
    else if (j >= K_TOP)   pick = K_TOP - 1;
    else {
      float dl = s_val[j - 1] - s;   // >= 0
      float dr = s - s_val[j];       // >= 0
      pick = (dl <= dr) ? (j - 1) : j;
    }
    orow[i] = s_idx[pick];
  }
}

// ---------------------------------------------------------------------------
// Launch
// ---------------------------------------------------------------------------
extern "C" void kernel_launch(void* const* d_in, const int* in_sizes, int n_in,
                              void* d_out, int out_size, void* d_ws, size_t ws_size,
                              hipStream_t stream) {
  const float* scores = (const float*)d_in[0];
  int* out = (int*)d_out;

  int B = in_sizes[0] / N_COLS;
  if (B < 1) B = 1;

  float* ref_val = (float*)d_ws;
  int*   ref_idx = (int*)((char*)d_ws + (size_t)B * K_TOP * sizeof(float));

  const size_t smemA = (size_t)CAP * 8 + 256 * sizeof(unsigned int) + 4 * sizeof(unsigned int);

  topk_radix_sort_kernel<<<dim3(B), dim3(TPB_A), smemA, stream>>>(scores, ref_val, ref_idx);
  nearest_gather_kernel<<<dim3(N_COLS / CHUNK, B), dim3(TPB_B), 0, stream>>>(
      scores, ref_val, ref_idx, out);
}